// DiverseBeamSearch_51221779972562
// MI455X (gfx1250) — compile-verified
//
#include <hip/hip_runtime.h>
#include <hip/hip_bf16.h>
#include <stdint.h>

// ---------------- configuration (matches reference constants) ----------------
#define CHUNKS 2     // row split for more parallelism in the heavy scan
#define TOPK   8     // k (=2) + max penalized positions (=6)
#define GG     4     // NUM_GROUPS
#define NGRAM  2     // N_GRAM
#define MB     2     // beam / NUM_GROUPS (and k = CAND_MULT*MB)
#define BLKA   256
#define NWAVE  (BLKA / 32)

// ---------------- CDNA5 async global->LDS path (guarded) ----------------
#if defined(__gfx1250__) && __has_builtin(__builtin_amdgcn_global_load_async_to_lds_b128)
#define HAVE_ASYNC_LDS 1
#else
#define HAVE_ASYNC_LDS 0
#endif

typedef int v4i __attribute__((vector_size(16)));

__device__ __forceinline__ void async_copy16(const float* g, float* l) {
#if HAVE_ASYNC_LDS
  // prototype: (v4i addrspace(1)* src, v4i addrspace(3)* dst, imm off, imm cpol)
  __builtin_amdgcn_global_load_async_to_lds_b128(
      (__attribute__((address_space(1))) v4i*)g,
      (__attribute__((address_space(3))) v4i*)l,
      /*imm offset*/0, /*cpol*/0);
#else
  float4 t = *(const float4*)g;
  *(float4*)l = t;
#endif
}

__device__ __forceinline__ void wait_async_le1() {
#if HAVE_ASYNC_LDS
#if __has_builtin(__builtin_amdgcn_s_wait_asynccnt)
  __builtin_amdgcn_s_wait_asynccnt(1);
#else
  asm volatile("s_wait_asynccnt 0x1" ::: "memory");
#endif
  asm volatile("" ::: "memory");
#endif
}

__device__ __forceinline__ void wait_async_0() {
#if HAVE_ASYNC_LDS
#if __has_builtin(__builtin_amdgcn_s_wait_asynccnt)
  __builtin_amdgcn_s_wait_asynccnt(0);
#else
  asm volatile("s_wait_asynccnt 0x0" ::: "memory");
#endif
  asm volatile("" ::: "memory");
#endif
}

// ---------------- top-8 helpers (strict key: larger value, then smaller idx) -
__device__ __forceinline__ bool keyGT(float v1, int e1, float v2, int e2) {
  return (v1 > v2) || (v1 == v2 && e1 < e2);
}

__device__ __forceinline__ void insert8(float (&tv)[TOPK], int (&ti)[TOPK],
                                        float v, int e) {
  if (!keyGT(v, e, tv[TOPK - 1], ti[TOPK - 1])) return;
  tv[TOPK - 1] = v; ti[TOPK - 1] = e;
#pragma unroll
  for (int q = TOPK - 1; q > 0; --q) {
    if (keyGT(tv[q], ti[q], tv[q - 1], ti[q - 1])) {
      float vt = tv[q]; tv[q] = tv[q - 1]; tv[q - 1] = vt;
      int   et = ti[q]; ti[q] = ti[q - 1]; ti[q - 1] = et;
    }
  }
}

// =====================================================================
// Kernel A: per (row, chunk) top-8 of raw lprobs, exact tie-breaking.
// One block per (row, chunk). Per-wave double-buffered async->LDS stream.
// =====================================================================
__global__ __launch_bounds__(BLKA)
void topk_rows_kernel(const float* __restrict__ lprobs,
                      float* __restrict__ wsv, int* __restrict__ wsi,
                      int vocab, int CL) {
  __shared__ __align__(16) float pipe[NWAVE][2][128];   // 32 lanes * 4 floats
  __shared__ float sv[BLKA * TOPK];
  __shared__ int   si[BLKA * TOPK];

  const int bx = blockIdx.x;
  const int r = bx / CHUNKS;
  const int c = bx % CHUNKS;
  const float* row = lprobs + (size_t)r * (size_t)vocab;
  const int c0 = c * CL;
  const int c1 = (c0 + CL < vocab) ? (c0 + CL) : vocab;

  const int tid  = threadIdx.x;
  const int w    = tid >> 5;
  const int lane = tid & 31;

  float tv[TOPK]; int ti[TOPK];
#pragma unroll
  for (int q = 0; q < TOPK; ++q) { tv[q] = -INFINITY; ti[q] = 0x7fffffff; }

  // 16B alignment of the global stream (row base is only 4B aligned)
  const long base_e = (long)r * (long)vocab;
  const int mis = (int)((base_e + c0) & 3);
  int a0 = c0 + ((4 - mis) & 3);
  if (a0 > c1) a0 = c1;
  const int nAligned4 = (c1 - a0) & ~3;
  const int t0 = a0 + nAligned4;

  // scalar head (<=3 elems) and tail (<=3 elems)
  if (tid < a0 - c0) { int e = c0 + tid; insert8(tv, ti, row[e], e); }
  if (tid < c1 - t0) { int e = t0 + tid; insert8(tv, ti, row[e], e); }

  if (nAligned4 >= 4) {
    const int safeE  = t0 - 4;                 // last valid 16B group start
    const int stride = NWAVE * 128;            // floats per block iteration
    const int ITERS  = (nAligned4 + stride - 1) / stride;

    auto issue = [&](int i) {
      int E  = a0 + (i * NWAVE + w) * 128 + lane * 4;
      int Ec = (E <= safeE) ? E : safeE;       // clamp: uniform issue per wave
      async_copy16(row + Ec, &pipe[w][i & 1][lane * 4]);
    };

    issue(0);
    for (int i = 0; i < ITERS; ++i) {
      if (i + 1 < ITERS) { issue(i + 1); wait_async_le1(); }
      else               { wait_async_0(); }
      int E = a0 + (i * NWAVE + w) * 128 + lane * 4;
      if (E <= safeE) {
        float4 d = *(const float4*)&pipe[w][i & 1][lane * 4];
        insert8(tv, ti, d.x, E);
        insert8(tv, ti, d.y, E + 1);
        insert8(tv, ti, d.z, E + 2);
        insert8(tv, ti, d.w, E + 3);
      }
    }
  }

  // block reduction: tree-merge of sorted-8 lists
#pragma unroll
  for (int q = 0; q < TOPK; ++q) { sv[tid * TOPK + q] = tv[q]; si[tid * TOPK + q] = ti[q]; }
  __syncthreads();
  for (int s = BLKA / 2; s > 0; s >>= 1) {
    if (tid < s) {
      const int A = tid * TOPK, B = (tid + s) * TOPK;
      float ov[TOPK]; int oi[TOPK];
      int x = 0, y = 0;
#pragma unroll
      for (int k2 = 0; k2 < TOPK; ++k2) {
        float va = sv[A + x]; int ea = si[A + x];
        float vb = sv[B + y]; int eb = si[B + y];
        bool take = keyGT(va, ea, vb, eb);
        ov[k2] = take ? va : vb;
        oi[k2] = take ? ea : eb;
        if (take) ++x; else ++y;
      }
#pragma unroll
      for (int k2 = 0; k2 < TOPK; ++k2) { sv[A + k2] = ov[k2]; si[A + k2] = oi[k2]; }
    }
    __syncthreads();
  }
  if (tid == 0) {
    int o = (r * CHUNKS + c) * TOPK;
#pragma unroll
    for (int q = 0; q < TOPK; ++q) { wsv[o + q] = sv[q]; wsi[o + q] = si[q]; }
  }
}

// =====================================================================
// Kernel B: per-batch sequential group logic (tiny). One thread per batch.
// =====================================================================
__global__ void select_kernel(const float* __restrict__ scores,
                              const float* __restrict__ go,
                              const int* __restrict__ obi,
                              const int* __restrict__ lng,
                              const int* __restrict__ mask,
                              const int* __restrict__ stepp,
                              const float* __restrict__ wsv,
                              const int* __restrict__ wsi,
                              float* __restrict__ out,
                              int bsz, int beam, int vocab, int T) {
  int b = blockIdx.x * blockDim.x + threadIdx.x;
  if (b >= bsz) return;

  const int step = stepp[0];
  const int ob   = obi[b];
  const int S    = bsz * beam;

  float sc[MB * GG]; bool valid[MB * GG];
  for (int j = 0; j < beam; ++j) {
    sc[j] = scores[((size_t)(b * beam + j)) * (size_t)T + (step - 1)];
    int sm = 0;
    for (int n = 0; n < NGRAM; ++n) sm += mask[(b * beam + j) * NGRAM + n];
    valid[j] = (sm == NGRAM);
  }

  int pidx[MB * GG]; int pg[MB * GG]; int cnt = 0;
  int idx_out[MB][GG];

  for (int g = 0; g < GG; ++g) {
    float bv[MB]; int bf[MB];
    for (int k = 0; k < MB; ++k) { bv[k] = -INFINITY; bf[k] = 0x7fffffff; }

    auto consider = [&](float v, int f) {
      for (int k = 0; k < MB; ++k) {
        if (v > bv[k] || (v == bv[k] && f < bf[k])) {
          for (int q = MB - 1; q > k; --q) { bv[q] = bv[q - 1]; bf[q] = bf[q - 1]; }
          bv[k] = v; bf[k] = f;
          break;
        }
      }
    };

    for (int i = 0; i < MB; ++i) {
      const int bm = g + GG * i;          // original beam index
      const float s = sc[bm];
      const int fb = i * vocab;           // flat base
      if (valid[bm]) {
        const int r = b * beam + bm;
        for (int c = 0; c < CHUNKS; ++c) {
          const int o = (r * CHUNKS + c) * TOPK;
          for (int q = 0; q < TOPK; ++q) {
            float v = wsv[o + q];
            int   e = wsi[o + q];
            if (e < 0 || e >= vocab) continue;
            float p = 0.0f;                // diversity penalty accumulator
            for (int t = 0; t < cnt; ++t)
              if (pidx[t] == e) p += 1.0f + go[(ob * GG + g) * GG + pg[t]];
            float val = (v + (-0.5f) * p) + s;   // DIVERSITY_STRENGTH = -0.5
            consider(val, fb + e);
          }
        }
      } else {
        // row overwritten to 0 then +sc: constant; ties -> smallest flat idx
        consider(s, fb + 0);
        consider(s, fb + 1);
      }
    }

    for (int k = 0; k < MB; ++k) {
      const int j = k * GG + g;           // output order: j = rank*G + g
      const int e = bf[k] % vocab;
      const int whichbeam = (bf[k] / vocab) * GG + g;
      out[b * beam + j]          = bv[k];
      out[S + b * beam + j]      = (float)e;
      out[2 * S + b * beam + j]  = (float)whichbeam;
      idx_out[k][g] = e;
      pidx[cnt] = e; pg[cnt] = g; ++cnt;
    }
  }

  // n-gram overlap update: new = (overlap + ov) * DIVERSITY_DISCOUNT
  const int O3 = 3 * S;
  for (int a = 0; a < GG; ++a) {
    for (int c2 = 0; c2 < GG; ++c2) {
      int ov = 0;
      for (int mbi = 0; mbi < MB; ++mbi) {
        const int ja = mbi * GG + a, jc = mbi * GG + c2;
        int match = 0;
        // n = 0: previous-token entries
        {
          bool ma = mask[(b * beam + ja) * NGRAM + 0] != 0;
          bool mc = mask[(b * beam + jc) * NGRAM + 0] != 0;
          int pa = lng[(b * beam + ja) * (NGRAM - 1) + 0];
          int pc = lng[(b * beam + jc) * (NGRAM - 1) + 0];
          if (ma && mc && pa == pc) ++match;
        }
        // n = 1: freshly selected indices
        {
          bool ma = mask[(b * beam + ja) * NGRAM + 1] != 0;
          bool mc = mask[(b * beam + jc) * NGRAM + 1] != 0;
          if (ma && mc && idx_out[mbi][a] == idx_out[mbi][c2]) ++match;
        }
        if (match == NGRAM) ++ov;
      }
      out[O3 + (ob * GG + a) * GG + c2] =
          (go[(ob * GG + a) * GG + c2] + (float)ov) * 0.5f;
    }
  }
}

// =====================================================================
// Launcher
// =====================================================================
extern "C" void kernel_launch(void* const* d_in, const int* in_sizes, int n_in,
                              void* d_out, int out_size, void* d_ws, size_t ws_size,
                              hipStream_t stream) {
  const float* lprobs = (const float*)d_in[0];
  const float* scores = (const float*)d_in[1];
  const float* go     = (const float*)d_in[2];
  const int*   obi    = (const int*)d_in[3];
  const int*   lng    = (const int*)d_in[4];
  const int*   mask   = (const int*)d_in[5];
  const int*   stepp  = (const int*)d_in[6];

  const int bsz  = in_sizes[3];
  const int beam = in_sizes[5] / (bsz * NGRAM);
  const int vocab = in_sizes[0] / (bsz * beam);
  const int T     = in_sizes[1] / (bsz * beam);
  const int rows  = bsz * beam;
  const int CL    = (vocab + CHUNKS - 1) / CHUNKS;

  float* wsv = (float*)d_ws;
  int*   wsi = (int*)((char*)d_ws + (size_t)rows * CHUNKS * TOPK * sizeof(float));

  topk_rows_kernel<<<dim3(rows * CHUNKS), dim3(BLKA), 0, stream>>>(
      lprobs, wsv, wsi, vocab, CL);

  const int nb = (bsz + 63) / 64;
  select_kernel<<<dim3(nb), dim3(64), 0, stream>>>(
      scores, go, obi, lng, mask, stepp, wsv, wsi, (float*)d_out,
      bsz, beam, vocab, T);
}